// Downsampling_77214922047594
// MI455X (gfx1250) — compile-verified
//
#include <hip/hip_runtime.h>
#include <hip/hip_bf16.h>

typedef float v2f __attribute__((ext_vector_type(2)));
typedef float v8f __attribute__((ext_vector_type(8)));

#define B_      4
#define N_      8192
#define M_      32768
#define KNN     16
#define C_      128
#define GROUPS_ 8
#define CPG     16
#define NEG_SLOPE 0.1f
#define EPS_GN  1e-5f
#define TOTAL_ROWS (B_ * N_)          // 32768
#define NSTATS  (3 * B_ * GROUPS_ * 2) // 192 floats
#define GN_COUNT 131072.0f            // N_ * CPG elements per (batch, group)

// ---------------------------------------------------------------------------
// Kernel 1: IDW gather-interpolation. One wave per query point.
// lanes 0..15 compute the 16 neighbor weights; each lane owns 4 channels.
// Block 0 also zero-inits the GroupNorm stats accumulators.
// ---------------------------------------------------------------------------
__global__ __launch_bounds__(256) void gather_idw_kernel(
    const float* __restrict__ s_feats,   // (B, M, C)
    const float* __restrict__ q_points,  // (B, N, 3)
    const float* __restrict__ s_points,  // (B, N, K, 3)
    const int*   __restrict__ nn_idx,    // (B, N, K)
    float* __restrict__ latent,          // (B*N, C)
    float* __restrict__ stats)           // NSTATS floats, zeroed here
{
    if (blockIdx.x == 0 && threadIdx.x < NSTATS) stats[threadIdx.x] = 0.0f;

    const int wave = threadIdx.x >> 5;
    const int lane = threadIdx.x & 31;
    const int p    = blockIdx.x * 8 + wave;   // global point id in [0, B*N)
    const int b    = p >> 13;                 // N_ == 8192

    // inverse-square-distance weight for neighbor k == lane (lanes 0..15)
    float w = 0.0f;
    if (lane < KNN) {
        const float qx = q_points[p * 3 + 0];
        const float qy = q_points[p * 3 + 1];
        const float qz = q_points[p * 3 + 2];
        const float* sp = s_points + ((size_t)p * KNN + lane) * 3;
        const float dx = sp[0] - qx, dy = sp[1] - qy, dz = sp[2] - qz;
        w = 1.0f / (dx * dx + dy * dy + dz * dz + 1e-8f);
    }
    float wsum = w;
#pragma unroll
    for (int m = 16; m >= 1; m >>= 1) wsum += __shfl_xor(wsum, m, 32);
    const float winv = 1.0f / wsum;

    const float* fb = s_feats + (size_t)b * M_ * C_;
    float4 acc = make_float4(0.f, 0.f, 0.f, 0.f);
#pragma unroll 4
    for (int k = 0; k < KNN; ++k) {
        const float wk = __shfl(w, k, 32) * winv;
        const int   id = nn_idx[(size_t)p * KNN + k];
        const float4 v = *(const float4*)(fb + (size_t)id * C_ + 4 * lane);
        acc.x += wk * v.x; acc.y += wk * v.y;
        acc.z += wk * v.z; acc.w += wk * v.w;
    }
    *(float4*)(latent + (size_t)p * C_ + 4 * lane) = acc;
}

// ---------------------------------------------------------------------------
// Kernel 2/3/4: fused [optional GN+LeakyReLU(+residual) on input] -> GEMM
// (128x128 W, bias) -> store pre-norm output + accumulate GN sum/sumsq per
// (batch, group) via atomics. 256 threads, 128 rows x 128 cols per block.
// Matrix core: v_wmma_f32_16x16x4_f32, K tiled by 4 over 128.
// ---------------------------------------------------------------------------
template <bool NORM_IN, bool RESID>
__global__ __launch_bounds__(256) void gemm_gn_kernel(
    const float* __restrict__ A_src,     // (32768, 128) input (pre-norm if NORM_IN)
    const float* __restrict__ stats_in,  // [b][g][2] raw sums for input GN
    const float* __restrict__ gamma_in,
    const float* __restrict__ beta_in,
    const float* __restrict__ resid,     // q_feats, if RESID
    const float* __restrict__ Wm,        // (128, 128) row-major (k, n)
    const float* __restrict__ bias,      // (128)
    float* __restrict__ out_pre,         // (32768, 128) pre-norm result
    float* __restrict__ stats_out)       // [b][g][2] accumulated here
{
    __shared__ float Wlds[64 * 256];     // K-pair swizzled: [(k>>1)][n][k&1]
    __shared__ float Alds[128 * 130];    // padded rows (bank-conflict relief)
    __shared__ float meanS[GROUPS_], istdS[GROUPS_];
    __shared__ float gamS[C_], betS[C_], biasS[C_];
    __shared__ float sstat[2 * GROUPS_];

    const int tid     = threadIdx.x;
    const int rowbase = blockIdx.x * 128;
    const int b       = rowbase >> 13;   // 8192 rows per batch, 128 | 8192

    if (tid < 2 * GROUPS_) sstat[tid] = 0.0f;
    if constexpr (NORM_IN) {
        if (tid < GROUPS_) {
            const float s = stats_in[b * 16 + tid * 2 + 0];
            const float q = stats_in[b * 16 + tid * 2 + 1];
            const float m = s / GN_COUNT;
            meanS[tid] = m;
            istdS[tid] = rsqrtf(q / GN_COUNT - m * m + EPS_GN);
        }
        if (tid < C_) { gamS[tid] = gamma_in[tid]; betS[tid] = beta_in[tid]; }
    }
    if (tid < C_) biasS[tid] = bias[tid];
    __syncthreads();

    // Stage weight matrix, swizzled so B-fragment pairs (k, k+1) are adjacent.
    for (int i = tid; i < C_ * C_; i += 256) {
        const int k = i >> 7, n = i & 127;
        Wlds[(k >> 1) * 256 + (n << 1) + (k & 1)] = Wm[i];
    }
    // Stage A tile, applying previous-block GroupNorm + LeakyReLU (+residual).
    for (int i = tid; i < 128 * C_; i += 256) {
        const int r = i >> 7, c = i & 127;
        float x = A_src[(size_t)(rowbase + r) * C_ + c];
        if constexpr (NORM_IN) {
            const int g = c >> 4;
            x = (x - meanS[g]) * istdS[g] * gamS[c] + betS[c];
            x = (x >= 0.0f) ? x : NEG_SLOPE * x;
            if constexpr (RESID) x += resid[(size_t)(rowbase + r) * C_ + c];
        }
        Alds[r * 130 + c] = x;
    }
    __syncthreads();

    const int wv  = tid >> 5;
    const int lane = tid & 31;
    const int l15 = lane & 15;
    const int hi  = lane >> 4;

    v8f acc[8];
#pragma unroll
    for (int n = 0; n < 8; ++n) {
        const float bv = biasS[n * 16 + l15];
#pragma unroll
        for (int j = 0; j < 8; ++j) acc[n][j] = bv;
    }

    // A frag (16x4 f32): lanes 0-15 hold K=k,k+1 ; lanes 16-31 hold K=k+2,k+3
    const float* Arow = &Alds[(wv * 16 + l15) * 130];
#pragma unroll 2
    for (int kk = 0; kk < 32; ++kk) {
        const v2f a = *(const v2f*)(Arow + kk * 4 + 2 * hi);
        const int kk2 = kk * 2 + hi;
        const float* wp = &Wlds[kk2 * 256 + l15 * 2];
#pragma unroll
        for (int n = 0; n < 8; ++n) {
            const v2f bf = *(const v2f*)(wp + n * 32);
            acc[n] = __builtin_amdgcn_wmma_f32_16x16x4_f32(
                false, a, false, bf, (short)0, acc[n], false, false);
        }
    }

    // Store pre-norm output + per-group partial sums (group == 16-col tile n).
    const int rb = rowbase + wv * 16 + hi * 8;   // C/D: VGPR j -> row j (+8 hi)
#pragma unroll
    for (int n = 0; n < 8; ++n) {
        float s = 0.0f, q = 0.0f;
#pragma unroll
        for (int j = 0; j < 8; ++j) {
            const float v = acc[n][j];
            out_pre[(size_t)(rb + j) * C_ + n * 16 + l15] = v;
            s += v; q += v * v;
        }
#pragma unroll
        for (int m = 16; m >= 1; m >>= 1) {
            s += __shfl_xor(s, m, 32);
            q += __shfl_xor(q, m, 32);
        }
        if (lane == 0) {
            atomicAdd(&sstat[n * 2 + 0], s);
            atomicAdd(&sstat[n * 2 + 1], q);
        }
    }
    __syncthreads();
    if (tid < 2 * GROUPS_) atomicAdd(&stats_out[b * 16 + tid], sstat[tid]);
}

// ---------------------------------------------------------------------------
// Kernel 5: in-place GroupNorm + LeakyReLU on h3_pre (already in d_out).
// ---------------------------------------------------------------------------
__global__ __launch_bounds__(256) void finalize_gn_kernel(
    float* __restrict__ h, const float* __restrict__ stats,
    const float* __restrict__ gamma, const float* __restrict__ beta)
{
    const int i  = blockIdx.x * 256 + threadIdx.x;  // float4 index, 2^20 total
    const int b  = i >> 18;                          // 262144 float4 per batch
    const int c4 = i & 31;
    const int c  = c4 * 4;
    const int g  = c >> 4;
    const float s = stats[b * 16 + g * 2 + 0];
    const float q = stats[b * 16 + g * 2 + 1];
    const float m = s / GN_COUNT;
    const float istd = rsqrtf(q / GN_COUNT - m * m + EPS_GN);
    float4 v = *(float4*)(h + (size_t)i * 4);
    float* vp = &v.x;
#pragma unroll
    for (int j = 0; j < 4; ++j) {
        float x = (vp[j] - m) * istd * gamma[c + j] + beta[c + j];
        vp[j] = (x >= 0.0f) ? x : NEG_SLOPE * x;
    }
    *(float4*)(h + (size_t)i * 4) = v;
}

// ---------------------------------------------------------------------------
extern "C" void kernel_launch(void* const* d_in, const int* in_sizes, int n_in,
                              void* d_out, int out_size, void* d_ws, size_t ws_size,
                              hipStream_t stream) {
    const float* q_feats  = (const float*)d_in[0];
    const float* s_feats  = (const float*)d_in[1];
    const float* q_points = (const float*)d_in[2];
    const float* s_points = (const float*)d_in[3];
    const int*   nn_idx   = (const int*)  d_in[4];
    const float* W1 = (const float*)d_in[5];
    const float* b1 = (const float*)d_in[6];
    const float* g1 = (const float*)d_in[7];
    const float* be1= (const float*)d_in[8];
    const float* W2 = (const float*)d_in[9];
    const float* b2 = (const float*)d_in[10];
    const float* g2 = (const float*)d_in[11];
    const float* be2= (const float*)d_in[12];
    const float* W3 = (const float*)d_in[13];
    const float* b3 = (const float*)d_in[14];
    const float* g3 = (const float*)d_in[15];
    const float* be3= (const float*)d_in[16];

    float* buf0  = (float*)d_ws;                        // latent, then h2_pre
    float* buf1  = buf0 + (size_t)TOTAL_ROWS * C_;      // h1_pre
    float* stats = buf1 + (size_t)TOTAL_ROWS * C_;      // 192 floats
    float* out   = (float*)d_out;                       // h3_pre, then final

    // 1) gather + IDW interpolation (also zeroes stats)
    gather_idw_kernel<<<TOTAL_ROWS / 8, 256, 0, stream>>>(
        s_feats, q_points, s_points, nn_idx, buf0, stats);
    // 2) h1_pre = latent @ W1 + b1 ; accumulate GN1 stats
    gemm_gn_kernel<false, false><<<TOTAL_ROWS / 128, 256, 0, stream>>>(
        buf0, nullptr, nullptr, nullptr, nullptr, W1, b1, buf1, stats + 0);
    // 3) h2_pre = leaky(gn1(h1_pre)) @ W2 + b2 ; GN2 stats
    gemm_gn_kernel<true, false><<<TOTAL_ROWS / 128, 256, 0, stream>>>(
        buf1, stats + 0, g1, be1, nullptr, W2, b2, buf0, stats + 64);
    // 4) h3_pre = (leaky(gn2(h2_pre)) + q_feats) @ W3 + b3 ; GN3 stats
    gemm_gn_kernel<true, true><<<TOTAL_ROWS / 128, 256, 0, stream>>>(
        buf0, stats + 64, g2, be2, q_feats, W3, b3, out, stats + 128);
    // 5) out = leaky(gn3(h3_pre)) in place
    finalize_gn_kernel<<<(TOTAL_ROWS * C_ / 4) / 256, 256, 0, stream>>>(
        out, stats + 128, g3, be3);
}